// GIN_90477781058260
// MI455X (gfx1250) — compile-verified
//
#include <hip/hip_runtime.h>

#define D_FEAT 128
#define GIN_EPS 0.1f

typedef float v2f __attribute__((ext_vector_type(2)));
typedef float v8f __attribute__((ext_vector_type(8)));

// Device-scope fp32 atomic add (no return) -> global_atomic_add_f32 at L2.
__device__ __forceinline__ void atomic_add_f32_dev(float* p, float v) {
    asm volatile("global_atomic_add_f32 %0, %1, off scope:SCOPE_DEV"
                 :: "v"(p), "v"(v) : "memory");
}

// Conflict-free pair-major LDS index for a 128x128 transposed weight matrix.
// Pair (k, k+1) stays contiguous (8B loads); at read time (k = 4*kk + 2*half)
// banks are 32*half + 2*r (mod 64): lanes 0-15 -> banks 0..31, lanes 16-31 ->
// banks 32..63 -> zero conflicts on the hot B-fragment reads.
__device__ __forceinline__ int w_idx(int k, int n) {
    return ((k >> 1) << 8) + ((2 * n + 32 * ((k >> 1) & 1)) & 255) + (k & 1);
}

// z[i] = (1 + eps) * h[i], float4-vectorized. Fully overwrites z.
__global__ __launch_bounds__(256) void gin_scale_kernel(
    const float* __restrict__ h, float* __restrict__ z, int n4) {
    int i = blockIdx.x * 256 + threadIdx.x;
    if (i >= n4) return;
    float4 v = ((const float4*)h)[i];
    v.x *= (1.0f + GIN_EPS);
    v.y *= (1.0f + GIN_EPS);
    v.z *= (1.0f + GIN_EPS);
    v.w *= (1.0f + GIN_EPS);
    ((float4*)z)[i] = v;
}

// One wave (32 lanes) per edge; each lane handles 4 contiguous channels.
// z[dst] += w * h[src] via device-scope f32 atomics (RMW at L2).
__global__ __launch_bounds__(256) void gin_scatter_kernel(
    const float* __restrict__ h, const long long* __restrict__ edge_index,
    const float* __restrict__ ew, float* __restrict__ z, int n_edges) {
    long long gid = (long long)blockIdx.x * 256 + threadIdx.x;
    int e = (int)(gid >> 5);
    int lane = (int)(gid & 31);
    if (e >= n_edges) return;
    long long src = edge_index[e];
    long long dst = edge_index[(long long)n_edges + e];
    float w = ew[e];
    float4 hv = *(const float4*)(h + src * D_FEAT + lane * 4);
    float* zp = z + dst * D_FEAT + lane * 4;
    atomic_add_f32_dev(zp + 0, w * hv.x);
    atomic_add_f32_dev(zp + 1, w * hv.y);
    atomic_add_f32_dev(zp + 2, w * hv.z);
    atomic_add_f32_dev(zp + 3, w * hv.w);
}

// out = relu(z @ Wa) @ Wb (optional relu on out), fp32 WMMA 16x16x4.
// 128 threads = 4 wave32; each wave owns a 16-row tile; grid-stride over
// 64-row chunks amortizes the 128KB weight staging across chunks.
// LDS: Wa 64KB + Wb 64KB (pair-major, conflict-free) + 32KB staging = 160KB.
__global__ __launch_bounds__(128) void gin_mlp_kernel(
    const float* __restrict__ z, const float* __restrict__ Wa,
    const float* __restrict__ Wb, float* __restrict__ out,
    int n_rows, int relu_out) {
    __shared__ float lWa[D_FEAT * D_FEAT];
    __shared__ float lWb[D_FEAT * D_FEAT];
    __shared__ float stage[4 * 16 * D_FEAT];

    const int tid = threadIdx.x;
    // Cooperative transposed weight stage (conflict-free: banks = 2n mod 64).
    for (int i = tid; i < D_FEAT * D_FEAT; i += 128) {
        int k = i >> 7;
        int n = i & (D_FEAT - 1);
        int w = w_idx(k, n);
        lWa[w] = Wa[i];
        lWb[w] = Wb[i];
    }
    __syncthreads();

    const int wave = tid >> 5;
    const int lane = tid & 31;
    const int half = lane >> 4;   // selects K pair / M high half
    const int r    = lane & 15;   // A/stage row, or B column-within-tile
    float* tw = stage + wave * (16 * D_FEAT);
    // Staging layout: word(row,c) = (c>>1)*32 + 2*row + (c&1)
    // -> A-fragment re-reads hit banks 32*half + 2*r : conflict-free.
    const int st_rd = half * 32 + 2 * r;          // + kk*64 + j
    const int st_wr = (r >> 1) * 32 + 16 * half + (r & 1);  // + nt*256 + 2*v

    const int nchunks = (n_rows + 63) >> 6;
    for (int chunk = blockIdx.x; chunk < nchunks; chunk += gridDim.x) {
        const int row0 = chunk * 64 + wave * 16;
        if (row0 >= n_rows) continue;  // wave-uniform: EXEC all-ones for WMMA

        int rowL = row0 + r;
        if (rowL >= n_rows) rowL = n_rows - 1;  // clamp loads; stores masked

        // ---- GEMM1: t = relu(z @ Wa) ----
        v2f afrag[32];
        {
            const float* zr = z + (size_t)rowL * D_FEAT + 2 * half;
            #pragma unroll
            for (int kk = 0; kk < 32; ++kk)
                afrag[kk] = *(const v2f*)(zr + kk * 4);
        }

        #pragma unroll 1
        for (int nt = 0; nt < 8; ++nt) {
            v8f c0 = {0.f, 0.f, 0.f, 0.f, 0.f, 0.f, 0.f, 0.f};
            v8f c1 = {0.f, 0.f, 0.f, 0.f, 0.f, 0.f, 0.f, 0.f};
            // Per-(nt,lane) base; kk advances by immediate offset only.
            const float* bp =
                lWa + half * 256 + ((32 * nt + 2 * r + 32 * half) & 255);
            #pragma unroll
            for (int kk = 0; kk < 16; ++kk) {
                v2f b0 = *(const v2f*)(bp + kk * 512);
                v2f b1 = *(const v2f*)(bp + (kk + 16) * 512);
                c0 = __builtin_amdgcn_wmma_f32_16x16x4_f32(
                         false, afrag[kk], false, b0, (short)0, c0, false, false);
                c1 = __builtin_amdgcn_wmma_f32_16x16x4_f32(
                         false, afrag[kk + 16], false, b1, (short)0, c1, false, false);
            }
            v8f c = c0 + c1;
            // C layout (M = v + 8*half, N = r) -> staging, with relu.
            #pragma unroll
            for (int v = 0; v < 8; ++v) {
                float val = c[v];
                val = val > 0.0f ? val : 0.0f;  // GIN internal relu
                tw[nt * 256 + st_wr + 2 * v] = val;
            }
        }

        // ---- GEMM2: out = t @ Wb ----
        v2f a2[32];
        {
            #pragma unroll
            for (int kk = 0; kk < 32; ++kk)
                a2[kk] = *(const v2f*)(tw + st_rd + kk * 64);
        }

        #pragma unroll 1
        for (int nt = 0; nt < 8; ++nt) {
            v8f c0 = {0.f, 0.f, 0.f, 0.f, 0.f, 0.f, 0.f, 0.f};
            v8f c1 = {0.f, 0.f, 0.f, 0.f, 0.f, 0.f, 0.f, 0.f};
            const float* bp =
                lWb + half * 256 + ((32 * nt + 2 * r + 32 * half) & 255);
            #pragma unroll
            for (int kk = 0; kk < 16; ++kk) {
                v2f b0 = *(const v2f*)(bp + kk * 512);
                v2f b1 = *(const v2f*)(bp + (kk + 16) * 512);
                c0 = __builtin_amdgcn_wmma_f32_16x16x4_f32(
                         false, a2[kk], false, b0, (short)0, c0, false, false);
                c1 = __builtin_amdgcn_wmma_f32_16x16x4_f32(
                         false, a2[kk + 16], false, b1, (short)0, c1, false, false);
            }
            v8f c = c0 + c1;
            #pragma unroll
            for (int v = 0; v < 8; ++v) {
                int rr = row0 + v + 8 * half;
                if (rr < n_rows) {
                    float val = c[v];
                    if (relu_out) val = val > 0.0f ? val : 0.0f;
                    out[(size_t)rr * D_FEAT + nt * 16 + r] = val;
                }
            }
        }
    }
}

extern "C" void kernel_launch(void* const* d_in, const int* in_sizes, int n_in,
                              void* d_out, int out_size, void* d_ws, size_t ws_size,
                              hipStream_t stream) {
    const float*     x   = (const float*)d_in[0];
    const long long* ei  = (const long long*)d_in[1];
    const float*     ew  = (const float*)d_in[2];
    const float*     W1a = (const float*)d_in[3];
    const float*     W1b = (const float*)d_in[4];
    const float*     W2a = (const float*)d_in[5];
    const float*     W2b = (const float*)d_in[6];
    float* out = (float*)d_out;

    const int n_nodes = in_sizes[0] / D_FEAT;
    const int n_edges = in_sizes[2];

    float* z = (float*)d_ws;                       // [N,128]
    float* h = z + (size_t)n_nodes * D_FEAT;       // [N,128]

    const int n4 = n_nodes * (D_FEAT / 4);
    const int scale_blocks = (n4 + 255) / 256;
    const long long sc_threads = (long long)n_edges * 32;
    const int scatter_blocks = (int)((sc_threads + 255) / 256);
    const int nchunks = (n_nodes + 63) / 64;
    const int mlp_blocks = (nchunks + 1) / 2;      // ~2 chunks per block

    // ---- Layer 1 ----
    gin_scale_kernel<<<scale_blocks, 256, 0, stream>>>(x, z, n4);
    gin_scatter_kernel<<<scatter_blocks, 256, 0, stream>>>(x, ei, ew, z, n_edges);
    gin_mlp_kernel<<<mlp_blocks, 128, 0, stream>>>(z, W1a, W1b, h, n_nodes, 1);

    // ---- Layer 2 ----
    gin_scale_kernel<<<scale_blocks, 256, 0, stream>>>(h, z, n4);
    gin_scatter_kernel<<<scatter_blocks, 256, 0, stream>>>(h, ei, ew, z, n_edges);
    gin_mlp_kernel<<<mlp_blocks, 128, 0, stream>>>(z, W2a, W2b, out, n_nodes, 0);
}